// GNN_36910948942080
// MI455X (gfx1250) — compile-verified
//
#include <hip/hip_runtime.h>
#include <hip/hip_bf16.h>

// ---------------------------------------------------------------------------
// GraphSAGE (3 layers, HID=64) + pool + MLP head for MI455X (gfx1250).
//  - Dense 64x64 GEMMs on V_WMMA_F32_16X16X4_F32 (full fp32: workload is
//    bandwidth bound, low-precision WMMA buys nothing).
//  - Per-block A-tiles (contiguous 16KB) streamed to LDS with
//    GLOBAL_LOAD_ASYNC_TO_LDS_B128 (ASYNCcnt).
//  - Weight tiles DMA'd by the Tensor Data Mover (TENSORcnt) via the 6-arg
//    tensor_load_to_lds builtin; async-b128 fallback otherwise.
//  - Scatter-mean via hardware float atomics; working set (~130MB) fits in
//    the 192MB L2 so atomics resolve on-chip.
// ---------------------------------------------------------------------------

typedef float v2f  __attribute__((ext_vector_type(2)));
typedef float v8f  __attribute__((ext_vector_type(8)));
typedef unsigned int u32x4 __attribute__((ext_vector_type(4)));
typedef int   i32x4 __attribute__((ext_vector_type(4)));
typedef int   i32x8 __attribute__((ext_vector_type(8)));

#define HID 64

// ---------------------------------------------------------------------------
// helpers: LDS byte offset of a __shared__ pointer (flat addr low 32 bits are
// the workgroup-relative LDS offset on AMDGPU), async copy, waits
// ---------------------------------------------------------------------------
__device__ __forceinline__ unsigned lds_off(const void* p) {
    return (unsigned)(size_t)p;
}

__device__ __forceinline__ void async_copy_b128(unsigned lds_byte_off,
                                                const void* gptr) {
    asm volatile("global_load_async_to_lds_b128 %0, %1, off"
                 :: "v"(lds_byte_off),
                    "v"((unsigned long long)(size_t)gptr)
                 : "memory");
}

__device__ __forceinline__ void wait_async0() {
    asm volatile("s_wait_asynccnt 0" ::: "memory");
}

__device__ __forceinline__ void wait_tensor0() {
#if __has_builtin(__builtin_amdgcn_s_wait_tensorcnt)
    __builtin_amdgcn_s_wait_tensorcnt(0);
#else
    asm volatile("s_wait_tensorcnt 0" ::: "memory");
#endif
}

#if __has_builtin(__builtin_amdgcn_tensor_load_to_lds)
#define SAGE_HAVE_TDM 1
// D# group 0: count=1 | lds_addr | global_addr[56:0] | type=2
__device__ __forceinline__ u32x4 tdm_g0(unsigned lds_byte_off, const void* gptr) {
    unsigned long long ga = (unsigned long long)(size_t)gptr;
    u32x4 g0;
    g0.x = 1u;                                   // count=1, flags 0
    g0.y = lds_byte_off;                         // lds_addr
    g0.z = (unsigned)(ga & 0xFFFFFFFFull);       // global_addr[31:0]
    g0.w = (unsigned)((ga >> 32) & 0x01FFFFFFull) | (2u << 30);  // [56:32]|type=2
    return g0;
}
// D# group 1: 2D tile of 4-byte elements
__device__ __forceinline__ i32x8 tdm_g1(unsigned tensor_d0, unsigned tensor_d1,
                                        unsigned tile_d0, unsigned tile_d1,
                                        unsigned long long stride0) {
    i32x8 g1;
    g1[0] = (int)(2u << 16);                                     // data_size=4B
    g1[1] = (int)((tensor_d0 & 0xFFFFu) << 16);                  // dim0[15:0]
    g1[2] = (int)(((tensor_d0 >> 16) & 0xFFFFu) |
                  ((tensor_d1 & 0xFFFFu) << 16));                // dim0 hi|dim1 lo
    g1[3] = (int)(((tensor_d1 >> 16) & 0xFFFFu) |
                  ((tile_d0 & 0xFFFFu) << 16));                  // dim1 hi|tile0
    g1[4] = (int)(tile_d1 & 0xFFFFu);                            // tile1
    g1[5] = (int)(unsigned)(stride0 & 0xFFFFFFFFull);            // stride0[31:0]
    g1[6] = (int)(unsigned)((stride0 >> 32) & 0xFFFFull);        // stride0[47:32]
    g1[7] = 0;
    return g1;
}
#endif

// ---------------------------------------------------------------------------
// degree / inverse degree
// ---------------------------------------------------------------------------
__global__ void degree_kernel(const int* __restrict__ dst, float* __restrict__ deg,
                              int n_edges) {
    int e = blockIdx.x * blockDim.x + threadIdx.x;
    if (e < n_edges) {
        unsafeAtomicAdd(&deg[dst[e]], 1.0f);
    }
}

__global__ void invdeg_kernel(const float* __restrict__ deg, float* __restrict__ idg,
                              int n_nodes) {
    int v = blockIdx.x * blockDim.x + threadIdx.x;
    if (v < n_nodes) {
        idg[v] = 1.0f / fmaxf(deg[v], 1.0f);
    }
}

// ---------------------------------------------------------------------------
// scatter-add of messages (edge gather + atomic accumulate at dst, in L2)
// ---------------------------------------------------------------------------
__global__ void scatter3_kernel(const float* __restrict__ x,
                                const int* __restrict__ src,
                                const int* __restrict__ dst,
                                float* __restrict__ agg3, int n_edges) {
    int e = blockIdx.x * blockDim.x + threadIdx.x;
    if (e < n_edges) {
        int s = src[e], d = dst[e];
        unsafeAtomicAdd(&agg3[d * 3 + 0], x[s * 3 + 0]);
        unsafeAtomicAdd(&agg3[d * 3 + 1], x[s * 3 + 1]);
        unsafeAtomicAdd(&agg3[d * 3 + 2], x[s * 3 + 2]);
    }
}

__global__ void scatter64_kernel(const float* __restrict__ h,
                                 const int* __restrict__ src,
                                 const int* __restrict__ dst,
                                 float* __restrict__ agg, long long n_work) {
    long long idx = (long long)blockIdx.x * blockDim.x + threadIdx.x;
    if (idx < n_work) {
        int e = (int)(idx >> 6);
        int f = (int)(idx & 63);
        unsafeAtomicAdd(&agg[(long long)dst[e] * HID + f],
                        h[(long long)src[e] * HID + f]);
    }
}

// ---------------------------------------------------------------------------
// layer 1: in_dim = 3 -> 64 (tiny K, plain VALU)
// ---------------------------------------------------------------------------
__global__ void layer1_kernel(const float* __restrict__ x,
                              const float* __restrict__ agg3,
                              const float* __restrict__ idg,
                              const float* __restrict__ W1l,
                              const float* __restrict__ b1l,
                              const float* __restrict__ W1r,
                              float* __restrict__ out, int n_nodes) {
    long long idx = (long long)blockIdx.x * blockDim.x + threadIdx.x;
    if (idx >= (long long)n_nodes * HID) return;
    int v = (int)(idx >> 6);
    int j = (int)(idx & 63);
    float scale = idg[v];
    float s = b1l[j];
#pragma unroll
    for (int k = 0; k < 3; ++k) {
        s += agg3[v * 3 + k] * scale * W1l[k * HID + j];
        s += x[v * 3 + k] * W1r[k * HID + j];
    }
    out[idx] = fmaxf(s, 0.0f);
}

// ---------------------------------------------------------------------------
// layers 2/3: out = relu( (agg*invdeg) @ Wl + bl + h @ Wr ), all dims 64.
// Block = 128 threads (4 waves) -> 64 output rows.
// Staging: Wl/Wr by TDM DMA, 64-row h/agg tiles by async-to-LDS b128.
// Compute: V_WMMA_F32_16X16X4_F32, one wave per 16x64 output tile.
// ---------------------------------------------------------------------------
__global__ void __launch_bounds__(128)
sage_gemm_wmma(const float* __restrict__ h,     // [N,64] self features
               const float* __restrict__ agg,   // [N,64] neighbor sums
               const float* __restrict__ idg,   // [N]
               const float* __restrict__ Wl,    // [64,64] (k-major)
               const float* __restrict__ bl,    // [64]
               const float* __restrict__ Wr,    // [64,64]
               float* __restrict__ out,         // [N,64]
               int n_nodes) {
    __shared__ float sWl[HID * HID];   // 16KB
    __shared__ float sWr[HID * HID];   // 16KB
    __shared__ float sH[64 * HID];     // 16KB: 64 rows of h
    __shared__ float sA[64 * HID];     // 16KB: 64 rows of agg

    const int wave = threadIdx.x >> 5;  // 0..3
    const int lane = threadIdx.x & 31;
    const int m    = lane & 15;
    const int half = lane >> 4;         // 0 or 1
    const int blk_row0 = blockIdx.x * 64;
    const int row0 = blk_row0 + wave * 16;

    // ---- weight staging: Tensor Data Mover (one wave issues the DMAs) ----
#if SAGE_HAVE_TDM
    if (threadIdx.x < 32) {
        i32x4 z4 = {0, 0, 0, 0};
        i32x8 z8 = {0, 0, 0, 0, 0, 0, 0, 0};
        i32x8 g1w = tdm_g1(/*tensor_d0=*/HID, /*tensor_d1=*/HID,
                           /*tile_d0=*/HID, /*tile_d1=*/HID, /*stride0=*/HID);
        __builtin_amdgcn_tensor_load_to_lds(tdm_g0(lds_off(sWl), Wl), g1w,
                                            z4, z4, z8, 0);
        __builtin_amdgcn_tensor_load_to_lds(tdm_g0(lds_off(sWr), Wr), g1w,
                                            z4, z4, z8, 0);
    }
#else
    {   // fallback: async b128 copies of the weights (1024 chunks each)
        for (int i = threadIdx.x; i < 1024; i += 128)
            async_copy_b128(lds_off(sWl) + i * 16, (const char*)Wl + i * 16);
        for (int i = threadIdx.x; i < 1024; i += 128)
            async_copy_b128(lds_off(sWr) + i * 16, (const char*)Wr + i * 16);
    }
#endif

    // ---- A-tile staging: contiguous 16KB per buffer, async to LDS ----
    {
        const long long max_elems = ((long long)n_nodes - blk_row0) * HID;
        const float* gH = h + (size_t)blk_row0 * HID;
        const float* gA = agg + (size_t)blk_row0 * HID;
#pragma unroll
        for (int it = 0; it < 8; ++it) {
            int i = threadIdx.x + it * 128;            // chunk of 4 floats
            long long elem = (long long)i * 4;
            long long s = (elem + 4 <= max_elems) ? elem : 0;  // clamp (guarded)
            async_copy_b128(lds_off(sH) + i * 16, gH + s);
            async_copy_b128(lds_off(sA) + i * 16, gA + s);
        }
    }

    wait_async0();
#if SAGE_HAVE_TDM
    if (threadIdx.x < 32) wait_tensor0();
#endif
    __syncthreads();

    // per-lane row (clamped so EXEC stays all-1s through the WMMAs)
    int row  = row0 + m;
    int rowc = row < n_nodes ? row : (n_nodes - 1);
    const float scale = idg[rowc];
    const int rloc = wave * 16 + m;     // row within the staged 64-row tile

    v8f acc[4];
#pragma unroll
    for (int jt = 0; jt < 4; ++jt)
#pragma unroll
        for (int i = 0; i < 8; ++i) acc[jt][i] = 0.0f;

    for (int k = 0; k < HID; k += 4) {
        const int ka = k + half * 2;
        v2f aAgg, aH;
        aAgg.x = sA[rloc * HID + ka]     * scale;
        aAgg.y = sA[rloc * HID + ka + 1] * scale;
        aH.x   = sH[rloc * HID + ka];
        aH.y   = sH[rloc * HID + ka + 1];
#pragma unroll
        for (int jt = 0; jt < 4; ++jt) {
            const int n = jt * 16 + (lane & 15);
            v2f bL, bR;
            bL.x = sWl[ka * HID + n];
            bL.y = sWl[(ka + 1) * HID + n];
            bR.x = sWr[ka * HID + n];
            bR.y = sWr[(ka + 1) * HID + n];
            acc[jt] = __builtin_amdgcn_wmma_f32_16x16x4_f32(
                false, aAgg, false, bL, (short)0, acc[jt], false, false);
            acc[jt] = __builtin_amdgcn_wmma_f32_16x16x4_f32(
                false, aH, false, bR, (short)0, acc[jt], false, false);
        }
    }

    // bias + relu + store (guard rows past n_nodes)
#pragma unroll
    for (int jt = 0; jt < 4; ++jt) {
        const int n = jt * 16 + (lane & 15);
        const float bias = bl[n];
#pragma unroll
        for (int i = 0; i < 8; ++i) {
            const int M = i + half * 8;
            const int r = row0 + M;
            if (r < n_nodes) {
                float v = acc[jt][i] + bias;
                out[(long long)r * HID + n] = fmaxf(v, 0.0f);
            }
        }
    }
}

// ---------------------------------------------------------------------------
// pooling: per-graph sum (for mean), max (uint trick: post-ReLU >= 0), count
// ---------------------------------------------------------------------------
__global__ void pool_kernel(const float* __restrict__ h,
                            const int* __restrict__ batch,
                            float* __restrict__ psum,
                            unsigned int* __restrict__ pmax,
                            float* __restrict__ pcnt, int n_nodes) {
    long long idx = (long long)blockIdx.x * blockDim.x + threadIdx.x;
    if (idx >= (long long)n_nodes * HID) return;
    int v = (int)(idx >> 6);
    int f = (int)(idx & 63);
    int b = batch[v];
    float val = h[idx];
    unsafeAtomicAdd(&psum[b * HID + f], val);
    atomicMax(&pmax[b * HID + f], __float_as_uint(val));  // val >= 0 after relu
    if (f == 0) unsafeAtomicAdd(&pcnt[b], 1.0f);
}

// ---------------------------------------------------------------------------
// head: g = [mean|max] (64x128); out = relu(g@Wh1 + bh1) @ Wh2 + bh2
// ---------------------------------------------------------------------------
__global__ void head_kernel(const float* __restrict__ psum,
                            const unsigned int* __restrict__ pmax,
                            const float* __restrict__ pcnt,
                            const float* __restrict__ Wh1,  // [128,64]
                            const float* __restrict__ bh1,  // [64]
                            const float* __restrict__ Wh2,  // [64,1]
                            const float* __restrict__ bh2,  // [1]
                            float* __restrict__ out) {
    __shared__ float red[64];
    const int b = blockIdx.x;
    const int j = threadIdx.x;
    const float inv_cnt = 1.0f / fmaxf(pcnt[b], 1.0f);
    float s = bh1[j];
#pragma unroll 4
    for (int k = 0; k < HID; ++k) {
        s += (psum[b * HID + k] * inv_cnt) * Wh1[k * HID + j];
    }
#pragma unroll 4
    for (int k = 0; k < HID; ++k) {
        s += __uint_as_float(pmax[b * HID + k]) * Wh1[(HID + k) * HID + j];
    }
    s = fmaxf(s, 0.0f);
    red[j] = s * Wh2[j];
    __syncthreads();
    for (int off = 32; off > 0; off >>= 1) {
        if (j < off) red[j] += red[j + off];
        __syncthreads();
    }
    if (j == 0) out[b] = red[0] + bh2[0];
}

// ---------------------------------------------------------------------------
// launch
// ---------------------------------------------------------------------------
extern "C" void kernel_launch(void* const* d_in, const int* in_sizes, int n_in,
                              void* d_out, int out_size, void* d_ws, size_t ws_size,
                              hipStream_t stream) {
    const float* x     = (const float*)d_in[0];
    const int*   ei    = (const int*)d_in[1];
    const int*   batch = (const int*)d_in[2];
    const float* W1l   = (const float*)d_in[3];
    const float* b1l   = (const float*)d_in[4];
    const float* W1r   = (const float*)d_in[5];
    const float* W2l   = (const float*)d_in[6];
    const float* b2l   = (const float*)d_in[7];
    const float* W2r   = (const float*)d_in[8];
    const float* W3l   = (const float*)d_in[9];
    const float* b3l   = (const float*)d_in[10];
    const float* W3r   = (const float*)d_in[11];
    const float* Wh1   = (const float*)d_in[12];
    const float* bh1   = (const float*)d_in[13];
    const float* Wh2   = (const float*)d_in[14];
    const float* bh2   = (const float*)d_in[15];

    const int N = in_sizes[0] / 3;   // nodes
    const int E = in_sizes[1] / 2;   // edges
    const int G = out_size;          // graphs (64)

    const int* src = ei;        // edge_index[0,:]
    const int* dst = ei + E;    // edge_index[1,:]

    // ---- workspace carve-up (floats) ----
    float* ws  = (float*)d_ws;
    float* hA  = ws;                          // N*64
    float* hB  = hA + (size_t)N * HID;        // N*64
    float* agg = hB + (size_t)N * HID;        // N*64 (layer1 uses first N*3)
    float* deg = agg + (size_t)N * HID;       // N
    float* idg = deg + N;                     // N
    float* psum = idg + N;                    // G*64
    unsigned int* pmax = (unsigned int*)(psum + (size_t)G * HID);  // G*64
    float* pcnt = (float*)(pmax + (size_t)G * HID);                // G

    const int T = 256;
    const long long NF = (long long)N * HID;
    const long long EF = (long long)E * HID;

    // ---- degree (shared by all layers) ----
    (void)hipMemsetAsync(deg, 0, (size_t)N * sizeof(float), stream);
    degree_kernel<<<(E + T - 1) / T, T, 0, stream>>>(dst, deg, E);
    invdeg_kernel<<<(N + T - 1) / T, T, 0, stream>>>(deg, idg, N);

    // ---- layer 1 (3 -> 64) ----
    (void)hipMemsetAsync(agg, 0, (size_t)N * 3 * sizeof(float), stream);
    scatter3_kernel<<<(E + T - 1) / T, T, 0, stream>>>(x, src, dst, agg, E);
    layer1_kernel<<<(int)((NF + T - 1) / T), T, 0, stream>>>(x, agg, idg, W1l, b1l,
                                                             W1r, hA, N);

    // ---- layer 2 (64 -> 64) ----
    (void)hipMemsetAsync(agg, 0, (size_t)NF * sizeof(float), stream);
    scatter64_kernel<<<(int)((EF + T - 1) / T), T, 0, stream>>>(hA, src, dst, agg, EF);
    sage_gemm_wmma<<<(N + 63) / 64, 128, 0, stream>>>(hA, agg, idg, W2l, b2l, W2r,
                                                      hB, N);

    // ---- layer 3 (64 -> 64) ----
    (void)hipMemsetAsync(agg, 0, (size_t)NF * sizeof(float), stream);
    scatter64_kernel<<<(int)((EF + T - 1) / T), T, 0, stream>>>(hB, src, dst, agg, EF);
    sage_gemm_wmma<<<(N + 63) / 64, 128, 0, stream>>>(hB, agg, idg, W3l, b3l, W3r,
                                                      hA, N);

    // ---- pooling + head ----
    (void)hipMemsetAsync(psum, 0, (size_t)G * HID * sizeof(float), stream);
    (void)hipMemsetAsync(pmax, 0, (size_t)G * HID * sizeof(unsigned int), stream);
    (void)hipMemsetAsync(pcnt, 0, (size_t)G * sizeof(float), stream);
    pool_kernel<<<(int)((NF + T - 1) / T), T, 0, stream>>>(hA, batch, psum, pmax,
                                                           pcnt, N);
    head_kernel<<<G, 64, 0, stream>>>(psum, pmax, pcnt, Wh1, bh1, Wh2, bh2,
                                      (float*)d_out);
}